// Trans_Gat_Layer_28724741276287
// MI455X (gfx1250) — compile-verified
//
#include <hip/hip_runtime.h>
#include <hip/hip_bf16.h>
#include <math.h>
#include <stdint.h>

// Problem constants (match reference)
#define Bn   32
#define Nn   128
#define FIN  384
#define Hh   384
#define EDIM 100
#define Vv   50
#define ALPHA 0.2f
#define NEGBIG -9.0e15f

#define USE_TDM 1

typedef __attribute__((ext_vector_type(2))) float v2f;
typedef __attribute__((ext_vector_type(8))) float v8f;
typedef __attribute__((ext_vector_type(4))) uint32_t u32x4;
typedef __attribute__((ext_vector_type(8))) int i32x8;

// ---------------------------------------------------------------------------
// TDM: stage a 2D tile (tile1 rows x tile0 contiguous f32) from global into
// LDS at byte offset lds_off, row stride `stride` (f32 units). D# per CDNA5
// CDNA5 ISA; groups 2/3 omitted (2D tensor).
// ---------------------------------------------------------------------------
__device__ __forceinline__ void tdm_load_2d(const void* gaddr, uint32_t lds_off,
                                            uint32_t tile0, uint32_t tile1,
                                            uint32_t td0, uint32_t td1,
                                            uint32_t stride) {
    uint64_t ga = (uint64_t)gaddr;
    u32x4 g0;
    g0.x = 1u;                                        // count=1, user mode
    g0.y = lds_off;                                   // lds_addr (bytes)
    g0.z = (uint32_t)ga;                              // global_addr[31:0]
    g0.w = (uint32_t)((ga >> 32) & 0x1FFFFFFu)        // global_addr[56:32]
         | (2u << 30);                                // type=2 ("image")
    i32x8 g1;
    g1[0] = (int)(2u << 16);                          // data_size=2 (4 bytes)
    g1[1] = (int)((td0 & 0xFFFFu) << 16);             // tensor_dim0[15:0]
    g1[2] = (int)(((td0 >> 16) & 0xFFFFu) | ((td1 & 0xFFFFu) << 16));
    g1[3] = (int)(((td1 >> 16) & 0xFFFFu) | (tile0 << 16)); // tile_dim0
    g1[4] = (int)(tile1 & 0xFFFFu);                   // tile_dim1 (tile_dim2=0)
    g1[5] = (int)stride;                              // tensor_dim0_stride[31:0]
    g1[6] = 0;
    g1[7] = 0;
    asm volatile("tensor_load_to_lds %0, %1" :: "s"(g0), "s"(g1) : "memory");
}

__device__ __forceinline__ void tdm_wait0() {
#if __has_builtin(__builtin_amdgcn_s_wait_tensorcnt)
    __builtin_amdgcn_s_wait_tensorcnt(0);
#else
    asm volatile("s_wait_tensorcnt 0x0" ::: "memory");
#endif
}

// Fallback cooperative staging: lds[i] = src[(i/tile0)*stride + i%tile0]
__device__ __forceinline__ void stage_panel_fallback(float* lds, const float* src,
                                                     uint32_t tile0, uint32_t tile1,
                                                     uint32_t stride) {
    for (uint32_t i = threadIdx.x; i < tile0 * tile1; i += blockDim.x)
        lds[i] = src[(i / tile0) * stride + (i % tile0)];
}

// ---------------------------------------------------------------------------
// Kernel 1: tiny precompute.
//  blocks 0..49 : ew_vocab[v] = leaky_relu( relu(tbl[v]@emw^T + emb) @ eww^T + ewb )
//  block  50    : S[t] = sum_{k in [128t,128t+128)} a1[k],  t=0..2
// ---------------------------------------------------------------------------
__global__ void prep_kernel(const float* __restrict__ tbl,
                            const float* __restrict__ emw,
                            const float* __restrict__ emb,
                            const float* __restrict__ eww,
                            const float* __restrict__ ewb,
                            const float* __restrict__ a,
                            float* __restrict__ ew_vocab,
                            float* __restrict__ Svals) {
    __shared__ float red[128];
    int tid = threadIdx.x;                // 128 threads
    if (blockIdx.x < Vv) {
        int v = blockIdx.x;
        float acc = 0.f;
        for (int m = tid; m < Hh; m += 128) {
            float s = emb[m];
            const float* tv = tbl + v * EDIM;
            const float* wm = emw + (size_t)m * EDIM;
            for (int d = 0; d < EDIM; ++d) s += tv[d] * wm[d];
            s = fmaxf(s, 0.f);            // relu
            acc += s * eww[m];
        }
        red[tid] = acc;
        __syncthreads();
        for (int off = 64; off > 0; off >>= 1) {
            if (tid < off) red[tid] += red[tid + off];
            __syncthreads();
        }
        if (tid == 0) {
            float e = red[0] + ewb[0];
            ew_vocab[v] = (e > 0.f) ? e : ALPHA * e;
        }
    } else {
        if (tid < 3) {
            float s = 0.f;
            for (int k = 0; k < 128; ++k) s += a[tid * 128 + k];
            Svals[tid] = s;
        }
    }
}

// ---------------------------------------------------------------------------
// Kernel 2a: h = x @ W   (M=4096, N=384, K=384, fp32)
// Block (8 waves) owns one n0 panel + 8 M-tiles. B panel TDM-staged to LDS as
// [k][16] (16 floats contiguous per k-row). All fragment reads are
// base + compile-time offset so DS/GLOBAL immediate offsets absorb addressing.
// WMMA fragments per ISA 7.12.2 (A: m=L&15, K-pair by L>>4; D: row r+8*(L>>4)).
// ---------------------------------------------------------------------------
__global__ void gemm_h_kernel(const float* __restrict__ x,
                              const float* __restrict__ W,
                              float* __restrict__ h) {
    __shared__ float ldsB[FIN * 16];                  // 24 KB, [k][n]
    int r    = blockIdx.x;                            // 768 blocks
    int tn   = r / 32;                                // 0..23
    int tmg  = r % 32;
    int n0   = tn * 16;
    int wv   = threadIdx.x >> 5;
    int lane = threadIdx.x & 31;
    int m0   = (tmg * 8 + wv) * 16;
    int l16  = lane & 15, khalf = lane >> 4;

#if USE_TDM
    if (wv == 0) {
        tdm_load_2d(W + n0, (uint32_t)(size_t)(void*)ldsB,
                    16u, (uint32_t)FIN, (uint32_t)Hh, (uint32_t)FIN, (uint32_t)Hh);
        tdm_wait0();
    }
#else
    stage_panel_fallback(ldsB, W + n0, 16u, (uint32_t)FIN, (uint32_t)Hh);
#endif
    __syncthreads();

    v8f acc = {};
    const float* abase = x + (size_t)(m0 + l16) * FIN + 2 * khalf;
    const float* bbase = ldsB + l16 + 32 * khalf;     // lds[(2*khalf)*16 + l16]
#pragma unroll 8
    for (int k = 0; k < FIN; k += 4) {
        v2f av; av.x = abase[k];      av.y = abase[k + 1];
        v2f bv; bv.x = bbase[k * 16]; bv.y = bbase[k * 16 + 16];
        acc = __builtin_amdgcn_wmma_f32_16x16x4_f32(false, av, false, bv,
                                                    (short)0, acc, false, false);
    }
    float* orow = h + (size_t)(m0 + 8 * khalf) * Hh + n0 + l16;
#pragma unroll
    for (int rr = 0; rr < 8; ++rr) orow[(size_t)rr * Hh] = acc[rr];
}

// ---------------------------------------------------------------------------
// Kernel 2b: xe = x @ etw^T   (B[k][n] = etw[n][k], k contiguous in memory)
// Panel staged as LDS [n][384] so each B fragment pair is one ds_load_b64.
// ---------------------------------------------------------------------------
__global__ void gemm_xe_kernel(const float* __restrict__ x,
                               const float* __restrict__ etw,
                               float* __restrict__ xe) {
    __shared__ float ldsB[16 * FIN];                  // 24 KB, [n][k]
    int r    = blockIdx.x;                            // 768 blocks
    int tn   = r / 32;
    int tmg  = r % 32;
    int n0   = tn * 16;
    int wv   = threadIdx.x >> 5;
    int lane = threadIdx.x & 31;
    int m0   = (tmg * 8 + wv) * 16;
    int l16  = lane & 15, khalf = lane >> 4;

#if USE_TDM
    if (wv == 0) {
        tdm_load_2d(etw + (size_t)n0 * FIN, (uint32_t)(size_t)(void*)ldsB,
                    (uint32_t)FIN, 16u, (uint32_t)FIN, (uint32_t)Hh, (uint32_t)FIN);
        tdm_wait0();
    }
#else
    stage_panel_fallback(ldsB, etw + (size_t)n0 * FIN, (uint32_t)FIN, 16u,
                         (uint32_t)FIN);
#endif
    __syncthreads();

    v8f acc = {};
    const float* abase = x + (size_t)(m0 + l16) * FIN + 2 * khalf;
    const float* bbase = ldsB + l16 * FIN + 2 * khalf;
#pragma unroll 8
    for (int k = 0; k < FIN; k += 4) {
        v2f av; av.x = abase[k]; av.y = abase[k + 1];
        v2f bv; bv.x = bbase[k]; bv.y = bbase[k + 1];  // ds_load_b64
        acc = __builtin_amdgcn_wmma_f32_16x16x4_f32(false, av, false, bv,
                                                    (short)0, acc, false, false);
    }
    float* orow = xe + (size_t)(m0 + 8 * khalf) * Hh + n0 + l16;
#pragma unroll
    for (int rr = 0; rr < 8; ++rr) orow[(size_t)rr * Hh] = acc[rr];
}

// ---------------------------------------------------------------------------
// Kernel 3: e2[b,j] = h[b,j,:] . a2  — one wave per row, lane-strided dot.
// ---------------------------------------------------------------------------
__global__ void e2_kernel(const float* __restrict__ h,
                          const float* __restrict__ a,
                          float* __restrict__ e2) {
    int gw = (blockIdx.x * blockDim.x + threadIdx.x) >> 5;   // 0..4095
    int lane = threadIdx.x & 31;
    const float* hr = h + (size_t)gw * Hh;
    const float* a2 = a + Hh;
    float s = 0.f;
    for (int m = lane; m < Hh; m += 32) s += hr[m] * a2[m];
    for (int off = 16; off > 0; off >>= 1) s += __shfl_xor(s, off, 32);
    if (lane == 0) e2[gw] = s;
}

// ---------------------------------------------------------------------------
// Kernel 4: att_text row softmax. One wave per (b,i) row; 4 columns per lane.
//  e[b,i,j] = leaky_relu(S0*h[b,i,3j] + S1*h[b,i,3j+1] + S2*h[b,i,3j+2] + e2[b,j])
// ---------------------------------------------------------------------------
__global__ void att_text_kernel(const float* __restrict__ h,
                                const int* __restrict__ adj,
                                const float* __restrict__ e2,
                                const float* __restrict__ Svals,
                                float* __restrict__ att_out) {
    int gw = (blockIdx.x * blockDim.x + threadIdx.x) >> 5;   // 0..4095
    int lane = threadIdx.x & 31;
    int b = gw >> 7, i = gw & 127;
    const float* hrow = h + ((size_t)b * Nn + i) * Hh;
    const int* adjrow = adj + ((size_t)b * Nn + i) * Nn;
    const float* e2b = e2 + b * Nn;
    float S0 = Svals[0], S1 = Svals[1], S2 = Svals[2];

    float vals[4];
    float mx = -INFINITY;
    for (int q = 0; q < 4; ++q) {
        int j = lane + 32 * q;
        float e = S0 * hrow[3 * j] + S1 * hrow[3 * j + 1] + S2 * hrow[3 * j + 2] + e2b[j];
        e = (e > 0.f) ? e : ALPHA * e;            // leaky relu (pre-mask, as in ref)
        e = (adjrow[j] > 0) ? e : NEGBIG;
        vals[q] = e;
        mx = fmaxf(mx, e);
    }
    for (int off = 16; off > 0; off >>= 1) mx = fmaxf(mx, __shfl_xor(mx, off, 32));
    float sum = 0.f;
    for (int q = 0; q < 4; ++q) { vals[q] = __expf(vals[q] - mx); sum += vals[q]; }
    for (int off = 16; off > 0; off >>= 1) sum += __shfl_xor(sum, off, 32);
    float inv = 1.f / sum;
    float* outrow = att_out + ((size_t)b * Nn + i) * Nn;
    for (int q = 0; q < 4; ++q) outrow[lane + 32 * q] = vals[q] * inv;
}

// ---------------------------------------------------------------------------
// Kernel 5: att_edge row softmax via 50-entry vocab LUT.
// ---------------------------------------------------------------------------
__global__ void att_edge_kernel(const int* __restrict__ dep,
                                const int* __restrict__ adj,
                                const float* __restrict__ ew_vocab,
                                float* __restrict__ att_out) {
    int gw = (blockIdx.x * blockDim.x + threadIdx.x) >> 5;
    int lane = threadIdx.x & 31;
    int b = gw >> 7, i = gw & 127;
    const int* deprow = dep + ((size_t)b * Nn + i) * Nn;
    const int* adjrow = adj + ((size_t)b * Nn + i) * Nn;

    float vals[4];
    float mx = -INFINITY;
    for (int q = 0; q < 4; ++q) {
        int j = lane + 32 * q;
        float e = ew_vocab[deprow[j]];
        e = (adjrow[j] > 0) ? e : NEGBIG;
        vals[q] = e;
        mx = fmaxf(mx, e);
    }
    for (int off = 16; off > 0; off >>= 1) mx = fmaxf(mx, __shfl_xor(mx, off, 32));
    float sum = 0.f;
    for (int q = 0; q < 4; ++q) { vals[q] = __expf(vals[q] - mx); sum += vals[q]; }
    for (int off = 16; off > 0; off >>= 1) sum += __shfl_xor(sum, off, 32);
    float inv = 1.f / sum;
    float* outrow = att_out + ((size_t)b * Nn + i) * Nn;
    for (int q = 0; q < 4; ++q) outrow[lane + 32 * q] = vals[q] * inv;
}

// ---------------------------------------------------------------------------
// Kernel 6: h_text = att_text @ h ; h_edge = att_edge @ xe
// per batch M=128, N=384, K=128, fp32 WMMA. Block owns one (gemm, b, n0)
// panel (TDM-staged) + all 8 M-tiles. Base+const-offset LDS reads.
// ---------------------------------------------------------------------------
__global__ void gemm_att_kernel(const float* __restrict__ att_text,
                                const float* __restrict__ att_edge,
                                const float* __restrict__ h,
                                const float* __restrict__ xe,
                                float* __restrict__ h_text,
                                float* __restrict__ h_edge) {
    __shared__ float ldsB[Nn * 16];                   // 8 KB: [k][n]
    int bid   = blockIdx.x;
    int which = bid / 768;                            // 0 -> text, 1 -> edge
    int r     = bid % 768;
    int b     = r / 24;
    int tn    = r % 24;
    int n0    = tn * 16;
    int wv    = threadIdx.x >> 5;
    int lane  = threadIdx.x & 31;
    int m0    = wv * 16;
    int l16   = lane & 15, khalf = lane >> 4;

    const float* A  = ((which == 0) ? att_text : att_edge) + (size_t)b * Nn * Nn;
    const float* Bm = ((which == 0) ? h : xe) + (size_t)b * Nn * Hh;
    float* out      = ((which == 0) ? h_text : h_edge) + (size_t)b * Nn * Hh;

#if USE_TDM
    if (wv == 0) {
        tdm_load_2d(Bm + n0, (uint32_t)(size_t)(void*)ldsB,
                    16u, (uint32_t)Nn, (uint32_t)Hh, (uint32_t)Nn, (uint32_t)Hh);
        tdm_wait0();
    }
#else
    stage_panel_fallback(ldsB, Bm + n0, 16u, (uint32_t)Nn, (uint32_t)Hh);
#endif
    __syncthreads();

    v8f acc = {};
    const float* abase = A + (size_t)(m0 + l16) * Nn + 2 * khalf;
    const float* bbase = ldsB + l16 + 32 * khalf;
#pragma unroll 8
    for (int k = 0; k < Nn; k += 4) {
        v2f av; av.x = abase[k];      av.y = abase[k + 1];
        v2f bv; bv.x = bbase[k * 16]; bv.y = bbase[k * 16 + 16];
        acc = __builtin_amdgcn_wmma_f32_16x16x4_f32(false, av, false, bv,
                                                    (short)0, acc, false, false);
    }
    float* orow = out + (size_t)(m0 + 8 * khalf) * Hh + n0 + l16;
#pragma unroll
    for (int rr = 0; rr < 8; ++rr) orow[(size_t)rr * Hh] = acc[rr];
}

// ---------------------------------------------------------------------------
extern "C" void kernel_launch(void* const* d_in, const int* in_sizes, int n_in,
                              void* d_out, int out_size, void* d_ws, size_t ws_size,
                              hipStream_t stream) {
    const float* x    = (const float*)d_in[0];   // (B,N,F_IN)
    const int*   adj  = (const int*)  d_in[1];   // (B,N,N)
    const int*   dep  = (const int*)  d_in[2];   // (B,N,N)
    const float* W    = (const float*)d_in[3];   // (F_IN,H)
    const float* a    = (const float*)d_in[4];   // (2H,1)
    const float* tbl  = (const float*)d_in[5];   // (V,E_DIM)
    const float* emw  = (const float*)d_in[6];   // (H,E_DIM)
    const float* emb  = (const float*)d_in[7];   // (H,)
    const float* eww  = (const float*)d_in[8];   // (1,H)
    const float* ewb  = (const float*)d_in[9];   // (1,)
    const float* etw  = (const float*)d_in[10];  // (H,F_IN)

    // Workspace layout (floats)
    float* ws = (float*)d_ws;
    const size_t n_h  = (size_t)Bn * Nn * Hh;    // 1,572,864
    float* h_buf  = ws;
    float* xe_buf = ws + n_h;
    float* e2_buf = ws + 2 * n_h;                // 4096
    float* S_buf  = e2_buf + (size_t)Bn * Nn;    // 3 (+pad)
    float* ewv    = S_buf + 4;                   // 50

    // Output layout: h_text | h_edge | att_text | att_edge
    float* out      = (float*)d_out;
    float* h_text   = out;
    float* h_edge   = out + n_h;
    float* att_text = out + 2 * n_h;
    float* att_edge = att_text + (size_t)Bn * Nn * Nn;

    // 1) vocab LUT + S sums
    prep_kernel<<<Vv + 1, 128, 0, stream>>>(tbl, emw, emb, eww, ewb, a, ewv, S_buf);

    // 2) h = x@W and xe = x@etw^T : 24 n-panels * 32 m-groups each
    gemm_h_kernel<<<768, 256, 0, stream>>>(x, W, h_buf);
    gemm_xe_kernel<<<768, 256, 0, stream>>>(x, etw, xe_buf);

    // 3) e2: one wave per (b,j) row
    e2_kernel<<<4096 / 8, 256, 0, stream>>>(h_buf, a, e2_buf);

    // 4,5) softmaxes: 4096 rows, 1 wave/row, 8 waves/block
    att_text_kernel<<<4096 / 8, 256, 0, stream>>>(h_buf, adj, e2_buf, S_buf, att_text);
    att_edge_kernel<<<4096 / 8, 256, 0, stream>>>(dep, adj, ewv, att_edge);

    // 6) h_text / h_edge : 2 gemms * 32 batches * 24 n-panels
    gemm_att_kernel<<<2 * 768, 256, 0, stream>>>(att_text, att_edge,
                                                 h_buf, xe_buf,
                                                 h_text, h_edge);
}